// GENConv_137438953767
// MI455X (gfx1250) — compile-verified
//
#include <hip/hip_runtime.h>

#define NN 50000
#define EE 800000
#define DD 64
#define EPSF 1e-7f

typedef __attribute__((ext_vector_type(2))) float v2f;
typedef __attribute__((ext_vector_type(8))) float v8f;

// ---------------------------------------------------------------------------
// Kernel 1: zero the per-node accumulators (segmax as +0.0f == u32 0, which is
// the identity for positive-float bit-pattern max; denom/numer as 0.0f).
// ---------------------------------------------------------------------------
__global__ void init_ws(float* __restrict__ segmax,
                        float* __restrict__ denom,
                        float* __restrict__ numer, int n) {
    int i = blockIdx.x * blockDim.x + threadIdx.x;
    if (i < n) {
        segmax[i] = 0.0f;
        denom[i]  = 0.0f;
        numer[i]  = 0.0f;
    }
}

// ---------------------------------------------------------------------------
// Kernel 2: per-(edge,feature) message + segment max via u32 atomicMax.
// m = relu(nf[src] + emb[attr]) + eps  >= eps > 0, so positive-float bit
// patterns order like unsigned ints.
// ---------------------------------------------------------------------------
__global__ void edge_max(const float* __restrict__ nf,
                         const int*   __restrict__ eattr,
                         const int*   __restrict__ src,
                         const int*   __restrict__ dst,
                         const float* __restrict__ emb,
                         float*       __restrict__ segmax) {
    long long gid = (long long)blockIdx.x * blockDim.x + threadIdx.x;
    if (gid >= (long long)EE * DD) return;
    int d = (int)(gid & (DD - 1));
    int e = (int)(gid >> 6);
    int s = src[e];
    int t = dst[e];
    int a = eattr[e];
    float m = nf[(long long)s * DD + d] + emb[a * DD + d];
    m = fmaxf(m, 0.0f) + EPSF;
    atomicMax((unsigned int*)(segmax + (long long)t * DD + d), __float_as_uint(m));
}

// ---------------------------------------------------------------------------
// Kernel 3: recompute m, z = exp(m - segmax[dst]); accumulate
// denom += z, numer += m*z  (f32 global atomics -> L2-resident accumulators).
// ---------------------------------------------------------------------------
__global__ void edge_sum(const float* __restrict__ nf,
                         const int*   __restrict__ eattr,
                         const int*   __restrict__ src,
                         const int*   __restrict__ dst,
                         const float* __restrict__ emb,
                         const float* __restrict__ segmax,
                         float*       __restrict__ denom,
                         float*       __restrict__ numer) {
    long long gid = (long long)blockIdx.x * blockDim.x + threadIdx.x;
    if (gid >= (long long)EE * DD) return;
    int d = (int)(gid & (DD - 1));
    int e = (int)(gid >> 6);
    int s = src[e];
    int t = dst[e];
    int a = eattr[e];
    float m = nf[(long long)s * DD + d] + emb[a * DD + d];
    m = fmaxf(m, 0.0f) + EPSF;
    long long ti = (long long)t * DD + d;
    float z = __expf(m - segmax[ti]);
    atomicAdd(denom + ti, z);
    atomicAdd(numer + ti, m * z);
}

// ---------------------------------------------------------------------------
// Kernel 4: feats = nf + numer/denom (guard empty segments), then
// out = feats @ W^T + b via V_WMMA_F32_16X16X4_F32.
// Block: 128 threads = 4 waves; 16-row tile; wave w owns column tile w.
// A-frag layout (16x4 f32): lanes 0-15 -> K=0,1 ; lanes 16-31 -> K=2,3.
// D layout (16x16 f32): VGPR v -> M = v + 8*(lane>=16), N = lane&15.
// ---------------------------------------------------------------------------
__global__ __launch_bounds__(128) void node_gemm(
        const float* __restrict__ nf,
        const float* __restrict__ numer,
        const float* __restrict__ denom,
        const float* __restrict__ W,
        const float* __restrict__ bias,
        float*       __restrict__ out) {
    __shared__ float fts[16 * 68];   // feats tile, padded stride 68
    __shared__ float Ws [64 * 68];   // W (row-major), padded stride 68

    const int tid     = threadIdx.x;
    const int rowbase = blockIdx.x * 16;   // N is an exact multiple of 16

    // Stage W: 4096 elems / 128 threads = 32 each (coalesced 64-wide rows).
    #pragma unroll
    for (int i = 0; i < 32; ++i) {
        int idx = tid + i * 128;
        int r = idx >> 6, c = idx & 63;
        Ws[r * 68 + c] = W[idx];
    }
    // Stage feats tile: 1024 elems / 128 threads = 8 each; fuse the
    // agg = numer/denom epilogue of the softmax aggregation here.
    #pragma unroll
    for (int i = 0; i < 8; ++i) {
        int idx = tid + i * 128;
        int r = idx >> 6, c = idx & 63;
        long long g = (long long)(rowbase + r) * DD + c;
        float den = denom[g];
        float agg = (den > 0.0f) ? (numer[g] / den) : 0.0f;
        fts[r * 68 + c] = nf[g] + agg;
    }
    __syncthreads();

    const int lane  = tid & 31;
    const int wave  = tid >> 5;      // column tile 0..3
    const int half  = lane >> 4;     // 0: K pair {0,1}; 1: K pair {2,3}
    const int m     = lane & 15;     // A: row M ; B/D: column N

    v8f acc = {};
    #pragma unroll
    for (int k0 = 0; k0 < DD; k0 += 4) {
        const int k = k0 + 2 * half;
        v2f a, b;
        // A[M=m][k], A[M=m][k+1]
        a.x = fts[m * 68 + k];
        a.y = fts[m * 68 + k + 1];
        // B[k][N=j] = W[j][k] with j = wave*16 + m  (out = feats @ W^T)
        b.x = Ws[(wave * 16 + m) * 68 + k];
        b.y = Ws[(wave * 16 + m) * 68 + k + 1];
        acc = __builtin_amdgcn_wmma_f32_16x16x4_f32(
                  /*neg_a=*/false, a, /*neg_b=*/false, b,
                  /*c_mod=*/(short)0, acc,
                  /*reuse_a=*/false, /*reuse_b=*/false);
    }

    const float bb = bias[wave * 16 + m];
    #pragma unroll
    for (int v = 0; v < 8; ++v) {
        int row = rowbase + v + 8 * half;
        out[(long long)row * DD + wave * 16 + m] = acc[v] + bb;
    }
}

// ---------------------------------------------------------------------------
extern "C" void kernel_launch(void* const* d_in, const int* in_sizes, int n_in,
                              void* d_out, int out_size, void* d_ws, size_t ws_size,
                              hipStream_t stream) {
    const float* nf    = (const float*)d_in[0];   // node_feats  [N*D]
    const int*   eattr = (const int*)  d_in[1];   // edge_attr   [E]
    const int*   src   = (const int*)  d_in[2];   // src         [E]
    const int*   dst   = (const int*)  d_in[3];   // dst         [E]
    const float* emb   = (const float*)d_in[4];   // emb_table   [NBOND*D]
    const float* W     = (const float*)d_in[5];   // W           [D*D]
    const float* bias  = (const float*)d_in[6];   // b           [D]
    float*       out   = (float*)d_out;           // [N*D] f32

    float* segmax = (float*)d_ws;
    float* denom  = segmax + (size_t)NN * DD;
    float* numer  = denom  + (size_t)NN * DD;     // total 38.4 MB of d_ws

    const int nd = NN * DD;                        // 3.2M
    init_ws<<<(nd + 255) / 256, 256, 0, stream>>>(segmax, denom, numer, nd);

    const long long tot = (long long)EE * DD;      // 51.2M (edge, feature) pairs
    const int eblocks = (int)((tot + 255) / 256);  // 200000 blocks
    edge_max<<<eblocks, 256, 0, stream>>>(nf, eattr, src, dst, emb, segmax);
    edge_sum<<<eblocks, 256, 0, stream>>>(nf, eattr, src, dst, emb, segmax,
                                          denom, numer);

    node_gemm<<<NN / 16, 128, 0, stream>>>(nf, numer, denom, W, bias, out);
}